// LIFLayer_19018115186897
// MI455X (gfx1250) — compile-verified
//
#include <hip/hip_runtime.h>
#include <cstdint>
#include <cstddef>

// LIF scan: out[n,t,c] = Vm after step t of
//   R  = H(Vm-1)*1 + H(R-1)*(R-1)
//   Vm = (w_in[c]*x[n,t,c] + (1-w_leak[c])*Vm) * H(-R)
// with H(v) = (v >= 0) ? 1 : 0, state initialized to zero.
//
// Shapes: B=32, G=4, T=512, C=256  ->  N = B*G = 128 sequences.
// Memory-bound streaming scan: use CDNA5 async global->LDS loads with
// triple-buffered chunks and per-wave ASYNCcnt waits for deep prefetch.

#define T_STEPS 512
#define C_CH    256
#define N_SEQ   128           // B*G
#define CPB     64            // channels per block (= blockDim.x, 2 waves)
#define TC      16            // timesteps per LDS chunk
#define NBUF    3             // triple buffering
#define NCHUNK  (T_STEPS / TC) // 32

// Defined for BOTH compile passes: the host pass parses (but never codegens)
// the kernel body, so the asm fallback is harmless there. On the device pass
// the builtin is preferred when the toolchain declares it.
#if __has_builtin(__builtin_amdgcn_s_wait_asynccnt)
#define WAIT_ASYNCCNT(n) __builtin_amdgcn_s_wait_asynccnt(n)
#else
#define WAIT_ASYNCCNT(n) asm volatile("s_wait_asynccnt %0" :: "n"(n) : "memory")
#endif

__device__ __forceinline__ void async_load_b32(unsigned lds_addr, unsigned goff,
                                               uint64_t sbase) {
  // GVS form: global addr = SADDR(64b SGPR pair) + VADDR(32b byte offset).
  // VDST carries the per-lane LDS byte address (low 32 bits of flat shared
  // pointer == LDS offset on gfx1250; hardware uses addr[31:0] for LDS).
  asm volatile("global_load_async_to_lds_b32 %0, %1, %2"
               :: "v"(lds_addr), "v"(goff), "s"(sbase)
               : "memory");
}

__global__ __launch_bounds__(CPB) void lif_scan_kernel(
    const float* __restrict__ x,
    const float* __restrict__ w_input,
    const float* __restrict__ w_leak,
    float* __restrict__ out)
{
  __shared__ float buf[NBUF][TC][CPB];   // 12 KB

  const int tid = threadIdx.x;           // 0..63
  const int blk = blockIdx.x;            // 0..511
  const int n   = blk >> 2;              // sequence 0..127
  const int c   = ((blk & 3) << 6) | tid; // channel 0..255

  const float win   = w_input[c];
  const float kleak = 1.0f - w_leak[c];

  const uint64_t xbase    = (uint64_t)(uintptr_t)x;
  const unsigned seq_goff = ((unsigned)n * (unsigned)(T_STEPS * C_CH)
                             + (unsigned)c) * 4u;          // byte offset of (n,0,c)
  const unsigned my_lds   = (unsigned)(uintptr_t)&buf[0][0][tid];

  // Issue one chunk (TC timesteps x 1 channel per lane) into buffer bi.
  auto issue = [&](int k, int bi) {
    const unsigned g0 = seq_goff + (unsigned)k * (unsigned)(TC * C_CH * 4);
    const unsigned l0 = my_lds + (unsigned)bi * (unsigned)(TC * CPB * 4);
#pragma unroll
    for (int t = 0; t < TC; ++t)
      async_load_b32(l0 + (unsigned)t * (CPB * 4u),
                     g0 + (unsigned)t * (C_CH * 4u), xbase);
  };

  // Prologue: fill the 3-deep pipeline (48 async ops in flight, < 63 max).
  issue(0, 0);
  issue(1, 1);
  issue(2, 2);

  float Vm = 0.0f, R = 0.0f;
  float* op = out + ((size_t)n * (size_t)(T_STEPS * C_CH) + (size_t)c);

  for (int k = 0; k < NCHUNK; ++k) {
    // Outstanding async ops at this point: chunks k, k+1, k+2 (those that
    // exist), 16 each. Loads complete in order, so waiting down to the count
    // owned by *future* chunks guarantees chunk k has fully landed in LDS.
    const int rem = NCHUNK - 1 - k;
    if (rem >= 2)      WAIT_ASYNCCNT(32);
    else if (rem == 1) WAIT_ASYNCCNT(16);
    else               WAIT_ASYNCCNT(0);

    const int bi = k % NBUF;
#pragma unroll
    for (int t = 0; t < TC; ++t) {
      const float xv = buf[bi][t][tid];          // ds_load_b32, conflict-free
      // R = H(Vm-1) + H(R-1)*(R-1)
      const float spike = ((Vm - 1.0f) >= 0.0f) ? 1.0f : 0.0f;
      const float rdec  = ((R  - 1.0f) >= 0.0f) ? (R - 1.0f) : 0.0f;
      R = spike + rdec;
      // Vm = (w_in*x + (1-w_leak)*Vm) * H(-R);  note H(-0) == 1
      const float v = __builtin_fmaf(kleak, Vm, win * xv);
      Vm = ((-R) >= 0.0f) ? v : 0.0f;
      __builtin_nontemporal_store(Vm, op + (size_t)(k * TC + t) * C_CH);
    }

    // Refill the buffer we just drained (data dependence through the stores
    // forces s_wait_dscnt before this point, so LDS reuse is safe).
    if (k + NBUF < NCHUNK) issue(k + NBUF, bi);
  }
}

extern "C" void kernel_launch(void* const* d_in, const int* in_sizes, int n_in,
                              void* d_out, int out_size, void* d_ws, size_t ws_size,
                              hipStream_t stream) {
  (void)in_sizes; (void)n_in; (void)out_size; (void)d_ws; (void)ws_size;
  const float* x  = (const float*)d_in[0];   // (B,G,T,C) f32
  const float* wi = (const float*)d_in[1];   // (C,)      f32
  const float* wl = (const float*)d_in[2];   // (C,)      f32
  float* out = (float*)d_out;                // (B,G,T,C) f32

  const int blocks = N_SEQ * (C_CH / CPB);   // 128 * 4 = 512
  lif_scan_kernel<<<dim3(blocks), dim3(CPB), 0, stream>>>(x, wi, wl, out);
}